// NaOp_27410481283133
// MI455X (gfx1250) — compile-verified
//
#include <hip/hip_runtime.h>

typedef __attribute__((ext_vector_type(2)))  float    v2f;
typedef __attribute__((ext_vector_type(8)))  float    v8f;

// ---------------- degree / norm kernels ----------------

__global__ __launch_bounds__(256) void k_zero_deg(unsigned int* __restrict__ deg, int n) {
    int i = blockIdx.x * 256 + threadIdx.x;
    if (i < n) deg[i] = 0u;
}

__global__ __launch_bounds__(256) void k_count_deg(const int* __restrict__ dst,
                                                   unsigned int* __restrict__ deg,
                                                   int n_edges) {
    int e = blockIdx.x * 256 + threadIdx.x;
    if (e < n_edges) atomicAdd(&deg[dst[e]], 1u);
}

__global__ __launch_bounds__(256) void k_dinv(const unsigned int* __restrict__ deg,
                                              float* __restrict__ dinv, int n) {
    int i = blockIdx.x * 256 + threadIdx.x;
    if (i < n) dinv[i] = rsqrtf((float)deg[i] + 1.0f);  // +1 self-loop, always >= 1
}

// ---------------- single-block exclusive scan: deg -> rowstart, and fill cursors ----
// n = 100K: each of 1024 threads serially sums ~98 elems, Hillis-Steele scan of the
// 1024 partials in LDS, then writes exclusive offsets. Microseconds; runs once.

__global__ __launch_bounds__(1024) void k_scan(const unsigned int* __restrict__ deg,
                                               unsigned int* __restrict__ rowstart,
                                               unsigned int* __restrict__ cursor, int n) {
    __shared__ unsigned int part[1024];
    const int t = threadIdx.x;
    const int chunk = (n + 1023) / 1024;
    int lo = t * chunk;
    int hi = lo + chunk; if (hi > n) hi = n; if (lo > n) lo = n;

    unsigned int s = 0;
    for (int i = lo; i < hi; ++i) s += deg[i];
    part[t] = s;
    __syncthreads();
    for (int off = 1; off < 1024; off <<= 1) {
        unsigned int v = (t >= off) ? part[t - off] : 0u;
        __syncthreads();
        part[t] += v;
        __syncthreads();
    }
    unsigned int run = (t == 0) ? 0u : part[t - 1];
    for (int i = lo; i < hi; ++i) {
        rowstart[i] = run;
        cursor[i]   = run;
        run += deg[i];
    }
    if (t == 1023) rowstart[n] = run;   // total = part[1023]
}

// ---------------- CSR fill: bucket src ids by destination ----------------

__global__ __launch_bounds__(256) void k_fill(const int* __restrict__ src,
                                              const int* __restrict__ dst,
                                              unsigned int* __restrict__ cursor,
                                              unsigned int* __restrict__ srclist,
                                              int n_edges) {
    int e = blockIdx.x * 256 + threadIdx.x;
    if (e < n_edges) {
        unsigned int pos = atomicAdd(&cursor[dst[e]], 1u);
        srclist[pos] = (unsigned int)src[e];
    }
}

// ---------------- fused WMMA GEMM ----------------
// Per block: 16 rows of x. 4 waves; wave w owns output columns [16w, 16w+16).
// h = x@W (workspace); out = x@W_lin + b_lin + b + dinv^2 * h  (self-loop folded in).

__global__ __launch_bounds__(128) void k_gemm_fused(
    const float* __restrict__ x,  const float* __restrict__ W,  const float* __restrict__ b,
    const float* __restrict__ Wl, const float* __restrict__ bl,
    const float* __restrict__ dinv,
    float* __restrict__ h, float* __restrict__ out, int n)
{
    const int lane = threadIdx.x & 31;
    const int wave = threadIdx.x >> 5;
    const int half = lane >> 4;
    const int rc   = lane & 15;
    const int row0 = blockIdx.x * 16;
    const int ncol = wave * 16 + rc;

    int arow = row0 + rc;
    if (arow >= n) arow = n - 1;

    v8f c  = {};
    v8f cl = {};

    // f32 WMMA: 16 K-steps of 4. A 16x4: lanes0-15 K={k0,k0+1}, lanes16-31 K={k0+2,k0+3}.
    for (int kt = 0; kt < 16; ++kt) {
        const int k0 = kt * 4 + half * 2;
        v2f a, bw, bw2;
        const float2 av = *(const float2*)(x + (size_t)arow * 64 + k0);
        a.x = av.x; a.y = av.y;
        bw.x  = W [(size_t)(k0    ) * 64 + ncol];
        bw.y  = W [(size_t)(k0 + 1) * 64 + ncol];
        bw2.x = Wl[(size_t)(k0    ) * 64 + ncol];
        bw2.y = Wl[(size_t)(k0 + 1) * 64 + ncol];
        c  = __builtin_amdgcn_wmma_f32_16x16x4_f32(false, a, false, bw,  (short)0, c,  false, false);
        cl = __builtin_amdgcn_wmma_f32_16x16x4_f32(false, a, false, bw2, (short)0, cl, false, false);
    }

    const float bias = b[ncol] + bl[ncol];
    #pragma unroll
    for (int r = 0; r < 8; ++r) {
        const int grow = row0 + r + 8 * half;
        if (grow < n) {
            const float hv = c[r];
            const float dv = dinv[grow];
            h  [(size_t)grow * 64 + ncol] = hv;
            out[(size_t)grow * 64 + ncol] = cl[r] + bias + dv * dv * hv;
        }
    }
}

// ---------------- gather-reduce (atomic-free aggregation) ----------------
// One 64-thread block per destination node; thread = feature column. Each
// iteration reads a contiguous 256B row of h (perfectly coalesced, L2-resident);
// dinv[dst] is factored out of the loop. Single non-atomic read-modify-write of out.

__global__ __launch_bounds__(64) void k_gather(const unsigned int* __restrict__ rowstart,
                                               const unsigned int* __restrict__ srclist,
                                               const float* __restrict__ h,
                                               const float* __restrict__ dinv,
                                               float* __restrict__ out, int n)
{
    const int d   = blockIdx.x;
    const int col = threadIdx.x;
    if (d >= n) return;
    const unsigned int beg = rowstart[d];
    const unsigned int end = rowstart[d + 1];
    float acc = 0.0f;
    for (unsigned int j = beg; j < end; ++j) {
        const unsigned int s = srclist[j];
        acc += h[(size_t)s * 64 + col] * dinv[s];
    }
    out[(size_t)d * 64 + col] += acc * dinv[d];
}

// ---------------- fallback: direct atomic scatter (if workspace too small) ----

__global__ __launch_bounds__(256) void k_scatter(const int* __restrict__ src,
                                                 const int* __restrict__ dst,
                                                 const float* __restrict__ h,
                                                 const float* __restrict__ dinv,
                                                 float* __restrict__ out, int n_edges)
{
    const long long t = (long long)blockIdx.x * 256 + threadIdx.x;
    const int e   = (int)(t >> 6);
    const int col = (int)(t & 63);
    if (e < n_edges) {
        const int s = src[e];
        const int d = dst[e];
        const float v = h[(size_t)s * 64 + col] * (dinv[s] * dinv[d]);
        unsafeAtomicAdd(out + (size_t)d * 64 + col, v);
    }
}

// ---------------- launch ----------------

extern "C" void kernel_launch(void* const* d_in, const int* in_sizes, int n_in,
                              void* d_out, int out_size, void* d_ws, size_t ws_size,
                              hipStream_t stream) {
    const float* x  = (const float*)d_in[0];
    const int*   ei = (const int*)  d_in[1];   // [2, E]: row0 = src, row1 = dst
    const float* W  = (const float*)d_in[2];
    const float* b  = (const float*)d_in[3];
    const float* Wl = (const float*)d_in[4];
    const float* bl = (const float*)d_in[5];
    float* out = (float*)d_out;

    const int n_nodes = in_sizes[0] / 64;
    const int n_edges = in_sizes[1] / 2;
    const int* src = ei;
    const int* dst = ei + n_edges;

    // workspace layout:
    //   h        [n*64 f32]
    //   deg      [n u32]      (doubles as CSR fill cursor after the scan)
    //   dinv     [n f32]
    //   rowstart [n+1 u32]
    //   srclist  [E u32]
    float*        h        = (float*)d_ws;
    unsigned int* deg      = (unsigned int*)(h + (size_t)n_nodes * 64);
    float*        dinv     = (float*)(deg + n_nodes);
    unsigned int* rowstart = (unsigned int*)(dinv + n_nodes);
    unsigned int* cursor   = deg;              // deg values consumed by the scan
    unsigned int* srclist  = rowstart + (n_nodes + 1);

    const size_t need_csr = ((size_t)n_nodes * 64 + 3 * (size_t)n_nodes + 1 + (size_t)n_edges) * 4;
    const bool use_csr = ws_size >= need_csr;

    const int nb_nodes = (n_nodes + 255) / 256;
    const int nb_edges = (n_edges + 255) / 256;

    k_zero_deg <<<nb_nodes, 256, 0, stream>>>(deg, n_nodes);
    k_count_deg<<<nb_edges, 256, 0, stream>>>(dst, deg, n_edges);
    k_dinv     <<<nb_nodes, 256, 0, stream>>>(deg, dinv, n_nodes);

    if (use_csr) {
        // scan consumes deg, then deg's storage becomes the fill cursor
        k_scan<<<1, 1024, 0, stream>>>(deg, rowstart, cursor, n_nodes);
        k_fill<<<nb_edges, 256, 0, stream>>>(src, dst, cursor, srclist, n_edges);
    }

    k_gemm_fused<<<(n_nodes + 15) / 16, 128, 0, stream>>>(x, W, b, Wl, bl, dinv, h, out, n_nodes);

    if (use_csr) {
        k_gather<<<n_nodes, 64, 0, stream>>>(rowstart, srclist, h, dinv, out, n_nodes);
    } else {
        const long long tot = (long long)n_edges * 64;
        k_scatter<<<(int)((tot + 255) / 256), 256, 0, stream>>>(src, dst, h, dinv, out, n_edges);
    }
}